// NeuralSemiLagrangian_78889959292955
// MI455X (gfx1250) — compile-verified
//
#include <hip/hip_runtime.h>
#include <math.h>

// Problem constants (B=1, C=256, H=180, W=360)
#define C_CH   256
#define H_DIM  180
#define W_DIM  360
#define HW     (H_DIM * W_DIM)      // 64800
#define O_DIM  (2 * C_CH)           // 512
#define HP     (H_DIM + 2)          // 182
#define WP     (W_DIM + 2)          // 362
#define PIX_PER_BLOCK 32
#define LDS_STRIDE 33               // row pad to dodge LDS bank conflicts

typedef __attribute__((ext_vector_type(2))) float v2f;
typedef __attribute__((ext_vector_type(8))) float v8f;

// ---------------------------------------------------------------------------
// Geo-cyclic padded fetch as index remapping (padded image never materialized)
// padded[Y, X], Y in [0,HP), X in [0,WP):
//   X==0 -> col W-1 ; X==W+1 -> col 0 ; else col = X-1
//   Y==0    -> row 0,   col = (col + W/2) % W
//   Y==H+1  -> row H-1, col = (col + W/2) % W
//   else    -> row Y-1
// ---------------------------------------------------------------------------
__device__ __forceinline__ float pad_fetch(const float* __restrict__ plane,
                                           int Y, int X)
{
    int col = (X == 0) ? (W_DIM - 1) : ((X == W_DIM + 1) ? 0 : (X - 1));
    int row;
    if (Y == 0) {
        row = 0;
        col = (col + W_DIM / 2) % W_DIM;
    } else if (Y == H_DIM + 1) {
        row = H_DIM - 1;
        col = (col + W_DIM / 2) % W_DIM;
    } else {
        row = Y - 1;
    }
    return plane[row * W_DIM + col];
}

__device__ __forceinline__ float cc1f(float x) {   // A = -0.75
    return (1.25f * x - 2.25f) * x * x + 1.0f;
}
__device__ __forceinline__ float cc2f(float x) {
    return ((-0.75f * x + 3.75f) * x - 6.0f) * x + 3.0f;
}

// ---------------------------------------------------------------------------
// Fused kernel: per block of 32 pixels
//   Phase 1: vel[512 x 32] = w_vel[512x256] x hidden[256 x 32pix] + bias
//            via V_WMMA_F32_16X16X4_F32, accumulated into LDS.
//   Phase 2: semi-Lagrangian departure points + bicubic resample for
//            256 channels x 32 pixels, u/v read from LDS.
// Grid: 2025 blocks x 256 threads (8 waves). Exact cover -> EXEC all ones.
// ---------------------------------------------------------------------------
__global__ __launch_bounds__(256) void nsl_fused(
    const float* __restrict__ hidden,    // [C, H, W]
    const float* __restrict__ lat_grid,  // [H, W]
    const float* __restrict__ lon_grid,  // [H, W]
    const float* __restrict__ w_vel,     // [2C, C]
    const float* __restrict__ b_vel,     // [2C]
    const int*   __restrict__ dtp,       // scalar
    float* __restrict__ out)             // [C, H, W]
{
    __shared__ float lds_vel[O_DIM * LDS_STRIDE];   // 512*33*4 = 67.5 KB

    const int lane = threadIdx.x & 31;
    const int wave = threadIdx.x >> 5;
    const int p0   = blockIdx.x * PIX_PER_BLOCK;    // 2025 * 32 = 64800 exact

    const int half = lane >> 4;                     // lane half selects K pair
    const int l    = lane & 15;

    // ---------------- Phase 1: WMMA GEMM into LDS ----------------
    // Wave owns M-tiles mt0..mt0+3 (each 16 rows of O) x 2 N-tiles (32 px).
    const int mt0 = wave * 4;

    v8f acc[4][2];
    #pragma unroll
    for (int j = 0; j < 4; ++j) {
        acc[j][0] = (v8f){0.f,0.f,0.f,0.f,0.f,0.f,0.f,0.f};
        acc[j][1] = (v8f){0.f,0.f,0.f,0.f,0.f,0.f,0.f,0.f};
    }

    #pragma unroll 2
    for (int k = 0; k < C_CH; k += 4) {
        // B fragments (4x16 fp32), shared by the wave's 4 M-tiles
        const float* bp = hidden + (size_t)(k + 2 * half) * HW + p0 + l;
        v2f b0, b1;
        b0.x = bp[0];        // K row k + 2*half
        b0.y = bp[HW];       // K row k + 2*half + 1
        b1.x = bp[16];
        b1.y = bp[16 + HW];
        #pragma unroll
        for (int j = 0; j < 4; ++j) {
            const int arow = (mt0 + j) * 16 + l;
            v2f a = *(const v2f*)(w_vel + (size_t)arow * C_CH + k + 2 * half);
            acc[j][0] = __builtin_amdgcn_wmma_f32_16x16x4_f32(
                false, a, false, b0, (short)0, acc[j][0], false, false);
            acc[j][1] = __builtin_amdgcn_wmma_f32_16x16x4_f32(
                false, a, false, b1, (short)0, acc[j][1], false, false);
        }
    }

    // C/D layout: VGPR r, lanes 0-15 -> M=r, lanes 16-31 -> M=8+r; N = lane&15
    #pragma unroll
    for (int j = 0; j < 4; ++j) {
        const int obase = (mt0 + j) * 16 + 8 * half;
        #pragma unroll
        for (int r = 0; r < 8; ++r) {
            const int o = obase + r;
            const float bias = b_vel[o];
            lds_vel[o * LDS_STRIDE + l]      = acc[j][0][r] + bias;
            lds_vel[o * LDS_STRIDE + 16 + l] = acc[j][1][r] + bias;
        }
    }

    __syncthreads();

    // ---------------- Phase 2: advection ----------------
    const int pix = threadIdx.x & 31;        // pixel within block
    const int p   = p0 + pix;                // global hw index
    const int cgrp = threadIdx.x >> 5;       // 0..7 -> channels cgrp*32 .. +31

    const float dtf  = (float)dtp[0];
    const float lat_p = lat_grid[p];
    const float lon_p = lon_grid[p];
    float slat, clat;
    sincosf(lat_p, &slat, &clat);            // pixel-only trig hoisted

    const float two_pi  = 6.2831853071795864769f;
    const float min_lat = -1.5697963267948966f;   // -pi/2 + 0.001
    const float max_lat =  1.5697963267948966f;   //  pi/2 - 0.001
    const float max_lon =  6.2657320146596423f;   // 2*pi*(1 - 1/360)
    const float sx = (float)W_DIM / (float)WP;
    const float sy = (float)H_DIM / (float)HP;

    #pragma unroll 1
    for (int i = 0; i < 32; ++i) {
        const int c = cgrp * 32 + i;         // wave-uniform channel
        const float u = lds_vel[c * LDS_STRIDE + pix];
        const float v = lds_vel[(C_CH + c) * LDS_STRIDE + pix];

        const float lon_prime = -u * dtf;
        const float lat_prime = -v * dtf;

        float slp, clp, slo, clo;
        sincosf(lat_prime, &slp, &clp);
        sincosf(lon_prime, &slo, &clo);

        float sin_lat = slp * clat + clp * clo * slat;
        sin_lat = fminf(fmaxf(sin_lat, -1.0f + 1e-07f), 1.0f - 1e-07f);
        const float lat_dep = asinf(sin_lat);

        const float num = clp * slo;
        const float den = clp * clo * clat - slp * slat;
        float xang = lon_p + atan2f(num, den) + two_pi;   // > 0
        const float lon_dep = fmodf(xang, two_pi);

        float grid_x = (2.0f * lon_dep / max_lon - 1.0f) * sx;
        float grid_y = (2.0f * (lat_dep - min_lat) / (max_lat - min_lat) - 1.0f) * sy;

        // wrap x into [-1,1): remainder(grid_x + 1, 2) - 1
        float gxs = grid_x + 1.0f;
        gxs = gxs - 2.0f * floorf(gxs * 0.5f);
        grid_x = gxs - 1.0f;
        // reflect y
        if (grid_y < -1.0f) grid_y = -(2.0f + grid_y);
        if (grid_y >  1.0f) grid_y =   2.0f - grid_y;

        const float ix = (grid_x + 1.0f) * 0.5f * (float)(WP - 1);
        const float iy = (grid_y + 1.0f) * 0.5f * (float)(HP - 1);
        const float ix0f = floorf(ix);
        const float iy0f = floorf(iy);
        const float tx = ix - ix0f;
        const float ty = iy - iy0f;
        const int ix0 = (int)ix0f;
        const int iy0 = (int)iy0f;

        float wxa[4], wya[4];
        wxa[0] = cc2f(tx + 1.0f); wxa[1] = cc1f(tx);
        wxa[2] = cc1f(1.0f - tx); wxa[3] = cc2f(2.0f - tx);
        wya[0] = cc2f(ty + 1.0f); wya[1] = cc1f(ty);
        wya[2] = cc1f(1.0f - ty); wya[3] = cc2f(2.0f - ty);

        const float* plane = hidden + (size_t)c * HW;
        float res = 0.0f;
        #pragma unroll
        for (int ii = 0; ii < 4; ++ii) {
            int yi = iy0 + (ii - 1);
            yi = yi < 0 ? 0 : (yi > HP - 1 ? HP - 1 : yi);
            float rowacc = 0.0f;
            #pragma unroll
            for (int jj = 0; jj < 4; ++jj) {
                int xj = ix0 + (jj - 1);
                xj = xj < 0 ? 0 : (xj > WP - 1 ? WP - 1 : xj);
                rowacc += wxa[jj] * pad_fetch(plane, yi, xj);
            }
            res += wya[ii] * rowacc;
        }
        out[(size_t)c * HW + p] = res;
    }
}

// ---------------------------------------------------------------------------
// Launcher
// ---------------------------------------------------------------------------
extern "C" void kernel_launch(void* const* d_in, const int* in_sizes, int n_in,
                              void* d_out, int out_size, void* d_ws, size_t ws_size,
                              hipStream_t stream) {
    const float* hidden   = (const float*)d_in[0];  // [1,256,180,360]
    const float* lat_grid = (const float*)d_in[1];  // [1,180,360]
    const float* lon_grid = (const float*)d_in[2];  // [1,180,360]
    const float* w_vel    = (const float*)d_in[3];  // [512,256]
    const float* b_vel    = (const float*)d_in[4];  // [512]
    const int*   dt       = (const int*)d_in[5];    // scalar
    float* out = (float*)d_out;                     // [1,256,180,360]

    // 64800 pixels / 32 per block = 2025 blocks, 256 threads (8 waves) each
    nsl_fused<<<2025, 256, 0, stream>>>(hidden, lat_grid, lon_grid,
                                        w_vel, b_vel, dt, out);
}